// QNN_10677288698276
// MI455X (gfx1250) — compile-verified
//
#include <hip/hip_runtime.h>
#include <hip/hip_bf16.h>

typedef __attribute__((ext_vector_type(2))) float v2f;
typedef __attribute__((ext_vector_type(8))) float v8f;

#define N_QUBITS 12
#define DIM 4096          // 2^12
#define N_LAYERS 2
#define OUT_FEATURES 32
#define N_BATCH 128

// One workgroup (256 threads = 8 wave32) simulates one circuit (o, b).
// State kept in LDS as separate re/im f32 arrays. RY layer applied as three
// 16x16 Kron-factor matrices via v_wmma_f32_16x16x4_f32 (exact f32 math).
__global__ __launch_bounds__(256) void QNN_sim_kernel(const float* __restrict__ x,
                                                      const float* __restrict__ w,
                                                      float* __restrict__ out) {
    __shared__ float sre[DIM];
    __shared__ float sim_[DIM];
    __shared__ float gate[3][256];   // gate[g][r*16+c] = G_g[r][c]
    __shared__ float red[256];

    const int tid = threadIdx.x;
    const int b = blockIdx.x;        // 0..127
    const int o = blockIdx.y;        // 0..31

    const int lane = tid & 31;
    const int waveId = tid >> 5;
    const int half = lane >> 4;      // 0: lanes 0-15, 1: lanes 16-31
    const int lr = lane & 15;

    // ---- initial state: (1/64) * exp(i*phi), phi = -0.5 * sum_q x_q * z_q(i)
    float xq[N_QUBITS];
#pragma unroll
    for (int q = 0; q < N_QUBITS; ++q) xq[q] = x[b * N_QUBITS + q];

#pragma unroll
    for (int t = 0; t < 16; ++t) {
        int i = t * 256 + tid;
        float s = 0.f;
#pragma unroll
        for (int q = 0; q < N_QUBITS; ++q) {
            float z = ((i >> (11 - q)) & 1) ? -1.f : 1.f;
            s += xq[q] * z;
        }
        float phi = -0.5f * s;
        sre[i] = 0.015625f * cosf(phi);
        sim_[i] = 0.015625f * sinf(phi);
    }

    for (int l = 0; l < N_LAYERS; ++l) {
        __syncthreads();
        // ---- build the three 16x16 Kron-factor gates for this layer
        {
            int r = tid >> 4, c = tid & 15;
#pragma unroll
            for (int g = 0; g < 3; ++g) {
                float p = 1.f;
#pragma unroll
                for (int t = 0; t < 4; ++t) {
                    int q = 4 * g + t;
                    float hw = 0.5f * w[o * (N_QUBITS * N_LAYERS) + q * N_LAYERS + l];
                    float cc = cosf(hw), ss = sinf(hw);
                    int rb = (r >> (3 - t)) & 1;
                    int cb = (c >> (3 - t)) & 1;
                    // RY = [[c,-s],[s,c]]
                    float m = rb ? (cb ? cc : ss) : (cb ? -ss : cc);
                    p *= m;
                }
                gate[g][tid] = p;
            }
        }
        __syncthreads();

        // ---- pass over axis k (bits 3..0, gate G2): rows=(i,j), D = X * G2^T
        for (int it = 0; it < 4; ++it) {
            int jb = waveId + 8 * it;           // 0..31
            int part = jb >> 4;                  // 0=re, 1=im
            int row0 = (jb & 15) * 16;
            float* S = part ? sim_ : sre;
            v8f d = {};
#pragma unroll
            for (int kk = 0; kk < 4; ++kk) {
                int kb = 4 * kk + 2 * half;
                v2f A, B;
                A.x = S[(row0 + lr) * 16 + kb + 0];       // X[m][k']
                A.y = S[(row0 + lr) * 16 + kb + 1];
                B.x = gate[2][lr * 16 + kb + 0];          // G2[n][k']
                B.y = gate[2][lr * 16 + kb + 1];
                d = __builtin_amdgcn_wmma_f32_16x16x4_f32(false, A, false, B,
                                                          (short)0, d, false, false);
            }
#pragma unroll
            for (int r = 0; r < 8; ++r)
                S[(row0 + r + 8 * half) * 16 + lr] = d[r];
        }
        __syncthreads();

        // ---- pass over axis j (bits 7..4, gate G1): D = G1 * slice_i
        for (int it = 0; it < 4; ++it) {
            int jb = waveId + 8 * it;
            int part = jb >> 4;
            int base = (jb & 15) * 256;
            float* S = part ? sim_ : sre;
            v8f d = {};
#pragma unroll
            for (int kk = 0; kk < 4; ++kk) {
                int kb = 4 * kk + 2 * half;
                v2f A, B;
                A.x = gate[1][lr * 16 + kb + 0];          // G1[j][j']
                A.y = gate[1][lr * 16 + kb + 1];
                B.x = S[base + (kb + 0) * 16 + lr];       // X[i][j'][k=n]
                B.y = S[base + (kb + 1) * 16 + lr];
                d = __builtin_amdgcn_wmma_f32_16x16x4_f32(false, A, false, B,
                                                          (short)0, d, false, false);
            }
#pragma unroll
            for (int r = 0; r < 8; ++r)
                S[base + (r + 8 * half) * 16 + lr] = d[r];
        }
        __syncthreads();

        // ---- pass over axis i (bits 11..8, gate G0): D = G0 * col_chunk
        for (int it = 0; it < 4; ++it) {
            int jb = waveId + 8 * it;
            int part = jb >> 4;
            int col0 = (jb & 15) * 16;
            float* S = part ? sim_ : sre;
            v8f d = {};
#pragma unroll
            for (int kk = 0; kk < 4; ++kk) {
                int kb = 4 * kk + 2 * half;
                v2f A, B;
                A.x = gate[0][lr * 16 + kb + 0];          // G0[i][i']
                A.y = gate[0][lr * 16 + kb + 1];
                B.x = S[(kb + 0) * 256 + col0 + lr];      // X[i'][col]
                B.y = S[(kb + 1) * 256 + col0 + lr];
                d = __builtin_amdgcn_wmma_f32_16x16x4_f32(false, A, false, B,
                                                          (short)0, d, false, false);
            }
#pragma unroll
            for (int r = 0; r < 8; ++r)
                S[(r + 8 * half) * 256 + col0 + lr] = d[r];
        }
        __syncthreads();

        // ---- CNOT chain as one composed permutation: new[i] = old[P(i)]
        // P = p_0∘p_1∘...∘p_10 ; p_q(j): flip bit(10-q) if bit(11-q) set
        float tre[16], tim[16];
#pragma unroll
        for (int t = 0; t < 16; ++t) {
            int i = t * 256 + tid;
            int j = i;
#pragma unroll
            for (int q = 10; q >= 0; --q) {
                if ((j >> (11 - q)) & 1) j ^= (1 << (10 - q));
            }
            tre[t] = sre[j];
            tim[t] = sim_[j];
        }
        __syncthreads();
#pragma unroll
        for (int t = 0; t < 16; ++t) {
            int i = t * 256 + tid;
            sre[i] = tre[t];
            sim_[i] = tim[t];
        }
    }
    __syncthreads();

    // ---- <Z_0> = sum_i (1 - 2*bit11(i)) * |psi_i|^2, deterministic tree reduce
    float part = 0.f;
#pragma unroll
    for (int t = 0; t < 16; ++t) {
        int i = t * 256 + tid;
        float re = sre[i], im = sim_[i];
        float p = re * re + im * im;
        part += ((i >> 11) & 1) ? -p : p;
    }
    red[tid] = part;
    __syncthreads();
    for (int s = 128; s > 0; s >>= 1) {
        if (tid < s) red[tid] += red[tid + s];
        __syncthreads();
    }
    if (tid == 0) out[o * N_BATCH + b] = red[0];
}

extern "C" void kernel_launch(void* const* d_in, const int* in_sizes, int n_in,
                              void* d_out, int out_size, void* d_ws, size_t ws_size,
                              hipStream_t stream) {
    const float* x = (const float*)d_in[0];       // [128, 12] f32
    const float* w = (const float*)d_in[1];       // [32, 12, 2] f32
    float* out = (float*)d_out;                   // [32, 128] f32
    (void)in_sizes; (void)n_in; (void)out_size; (void)d_ws; (void)ws_size;
    dim3 grid(N_BATCH, OUT_FEATURES);
    QNN_sim_kernel<<<grid, 256, 0, stream>>>(x, w, out);
}